// RegionContrast_90752658964656
// MI455X (gfx1250) — compile-verified
//
#include <hip/hip_runtime.h>
#include <hip/hip_bf16.h>

// ---------------- problem constants ----------------
#define BS        16
#define C_DIM     256
#define K_CLS     6
#define HW        16384          // 128*128
#define Q_DIM     4096
#define BLOCK_PIX 256            // pixels per label block (one WG pass)
#define NBLK      ((BS * HW) / BLOCK_PIX)   // 1024
#define INV_T     5.0f           // 1 / 0.2
#define NORM_EPS  1e-12f

typedef __attribute__((ext_vector_type(2))) float v2f;
typedef __attribute__((ext_vector_type(8))) float v8f;

// ---------------- init: zero accumulators + output ----------------
__global__ void rc_init(float* __restrict__ acc, float* __restrict__ out) {
    int i = blockIdx.x * blockDim.x + threadIdx.x;
    if (i < C_DIM * K_CLS + K_CLS) acc[i] = 0.0f;   // sums[6][256] + counts[6]
    if (i == 0) out[0] = 0.0f;
}

// ---------------- stage A: per-class masked sums via V_WMMA_F32_16X16X4_F32 ----
// sums[k][c] = sum over pixels with argmax(pred)==k of fea[b,c,p]
// A tile = 16 channels x 4 pixels (f32), B = onehot 4 pixels x 16 classes.
__global__ void rc_stageA(const float* __restrict__ fea,
                          const float* __restrict__ pred,
                          float* __restrict__ sums,      // [K_CLS][C_DIM]
                          float* __restrict__ counts) {  // [K_CLS]
    __shared__ int   s_lab[BLOCK_PIX];
    __shared__ float s_cnt[K_CLS];

    const int tid  = threadIdx.x;
    const int lane = tid & 31;
    const int wave = tid >> 5;              // 0..7
    const int t0   = wave * 2;              // channel tile 0 of this wave
    const int t1   = wave * 2 + 1;          // channel tile 1
    const int m    = lane & 15;             // A row (channel-in-tile) == B col (class)
    const int koff = (lane >> 4) * 2;       // pixel sub-offset: lanes16-31 handle K=2,3

    if (tid < K_CLS) s_cnt[tid] = 0.0f;

    v8f acc0 = {};   // D = 16 ch x 16 classes (f32)
    v8f acc1 = {};

    for (int blk = blockIdx.x; blk < NBLK; blk += gridDim.x) {
        const int b    = blk / (HW / BLOCK_PIX);
        const int pblk = (blk % (HW / BLOCK_PIX)) * BLOCK_PIX;

        __syncthreads();   // protect s_lab/s_cnt across block iterations
        {
            // one label per thread for this 256-pixel block
            const int p = pblk + tid;
            const float* pp = pred + ((long)b * K_CLS) * HW + p;
            float best = pp[0];
            int   bi   = 0;
            #pragma unroll
            for (int k = 1; k < K_CLS; ++k) {
                float v = pp[(long)k * HW];
                if (v > best) { best = v; bi = k; }   // first-max tie-break (argmax)
            }
            s_lab[tid] = bi;
            atomicAdd(&s_cnt[bi], 1.0f);
        }
        __syncthreads();

        const float* f0 = fea + ((long)b * C_DIM + t0 * 16 + m) * HW + pblk + koff;
        const float* f1 = fea + ((long)b * C_DIM + t1 * 16 + m) * HW + pblk + koff;

        #pragma unroll 4
        for (int g = 0; g < BLOCK_PIX / 4; ++g) {
            // A fragments: VGPR j holds pixel K = j + koff  (ISA 16x4 f32 A layout)
            v2f a0 = *(const v2f*)(f0 + 4 * g);
            v2f a1 = *(const v2f*)(f1 + 4 * g);
            // B fragment (onehot): VGPR j holds row K = j + koff, col N = m
            int lab0 = s_lab[4 * g + koff];
            int lab1 = s_lab[4 * g + koff + 1];
            v2f bm;
            bm.x = (lab0 == m) ? 1.0f : 0.0f;
            bm.y = (lab1 == m) ? 1.0f : 0.0f;

            acc0 = __builtin_amdgcn_wmma_f32_16x16x4_f32(
                       false, a0, false, bm, (short)0, acc0, false, false);
            acc1 = __builtin_amdgcn_wmma_f32_16x16x4_f32(
                       false, a1, false, bm, (short)0, acc1, false, false);
        }
    }

    // Flush D: value (M = r + 8*(lane>=16), N = m). Only classes 0..5 are real.
    if (m < K_CLS) {
        const int rofs = 8 * (lane >> 4);
        #pragma unroll
        for (int r = 0; r < 8; ++r) {
            atomicAdd(&sums[m * C_DIM + t0 * 16 + r + rofs], acc0[r]);
            atomicAdd(&sums[m * C_DIM + t1 * 16 + r + rofs], acc1[r]);
        }
    }
    __syncthreads();
    if (tid < K_CLS) atomicAdd(&counts[tid], s_cnt[tid]);
}

// ---------------- stage B: keys = normalize(sums / max(count,1)) -------------
__global__ void rc_stageB(const float* __restrict__ sums,
                          const float* __restrict__ counts,
                          float* __restrict__ keys) {
    __shared__ float red[256];
    const int c = threadIdx.x;
    for (int k = 0; k < K_CLS; ++k) {
        float cnt = counts[k];
        float v   = sums[k * C_DIM + c] / fmaxf(cnt, 1.0f);
        red[c] = v * v;
        __syncthreads();
        for (int s = 128; s > 0; s >>= 1) {
            if (c < s) red[c] += red[c + s];
            __syncthreads();
        }
        float nrm = fmaxf(sqrtf(red[0]), NORM_EPS);
        __syncthreads();
        keys[k * C_DIM + c] = v / nrm;
    }
}

// ---------------- stage D: per-(k,c) row log-softmax, loss accumulation ------
// One block per channel c handles all 6 classes; exactly one pass over queues.
__global__ void rc_stageD(const float* __restrict__ queues,  // [K][C][Q]
                          const float* __restrict__ keys,    // [K][C]
                          const float* __restrict__ counts,  // [K]
                          float* __restrict__ out) {
    const int c   = blockIdx.x;
    const int tid = threadIdx.x;
    __shared__ float s_m[256];
    __shared__ float s_s[256];

    float key[K_CLS];
    #pragma unroll
    for (int k = 0; k < K_CLS; ++k) key[k] = keys[k * C_DIM + c] * INV_T;

    float mk[K_CLS], sk[K_CLS];
    #pragma unroll
    for (int k = 0; k < K_CLS; ++k) { mk[k] = -3.0e38f; sk[k] = 0.0f; }

    for (int j = tid; j < Q_DIM; j += 256) {
        float qv[K_CLS];
        float qt = 0.0f;
        #pragma unroll
        for (int k = 0; k < K_CLS; ++k) {
            qv[k] = queues[((long)(k * C_DIM + c)) * Q_DIM + j];
            qt += qv[k];
        }
        #pragma unroll
        for (int k = 0; k < K_CLS; ++k) {
            float x1 = key[k] * qv[k];          // l_pos logit
            float x2 = key[k] * (qt - qv[k]);   // l_neg logit
            float mx = fmaxf(x1, x2);
            float mn = fmaxf(mk[k], mx);
            sk[k] = sk[k] * __expf(mk[k] - mn) + __expf(x1 - mn) + __expf(x2 - mn);
            mk[k] = mn;
        }
    }

    float total = 0.0f;   // only tid 0 uses it
    for (int k = 0; k < K_CLS; ++k) {
        s_m[tid] = mk[k];
        s_s[tid] = sk[k];
        __syncthreads();
        for (int str = 128; str > 0; str >>= 1) {
            if (tid < str) {
                float m2 = s_m[tid + str], s2 = s_s[tid + str];
                float M  = fmaxf(s_m[tid], m2);
                s_s[tid] = s_s[tid] * __expf(s_m[tid] - M) + s2 * __expf(m2 - M);
                s_m[tid] = M;
            }
            __syncthreads();
        }
        if (tid == 0) {
            float q0    = queues[((long)(k * C_DIM + c)) * Q_DIM];
            float x0    = key[k] * q0;                       // logits[...,0]
            float logp0 = x0 - (s_m[0] + __logf(s_s[0]));
            if (counts[k] > 0.0f) total += -logp0 * (1.0f / (float)C_DIM);
        }
        __syncthreads();
    }
    if (tid == 0) atomicAdd(out, total);
}

// ---------------- launcher ----------------
extern "C" void kernel_launch(void* const* d_in, const int* in_sizes, int n_in,
                              void* d_out, int out_size, void* d_ws, size_t ws_size,
                              hipStream_t stream) {
    const float* fea    = (const float*)d_in[0];   // [16,256,128,128]
    const float* pred   = (const float*)d_in[1];   // [16,6,128,128]
    const float* queues = (const float*)d_in[2];   // [6,256,4096]
    float* out = (float*)d_out;
    float* ws  = (float*)d_ws;

    float* sums   = ws;                        // 1536 floats
    float* counts = ws + K_CLS * C_DIM;        // 6 floats
    float* keys   = counts + 8;                // 1536 floats (8-aligned)

    rc_init<<<7, 256, 0, stream>>>(sums, out);
    rc_stageA<<<256, 256, 0, stream>>>(fea, pred, sums, counts);
    rc_stageB<<<1, 256, 0, stream>>>(sums, counts, keys);
    rc_stageD<<<C_DIM, 256, 0, stream>>>(queues, keys, counts, out);
}